// GLMBlockInputTransform_25202868093664
// MI455X (gfx1250) — compile-verified
//
#include <hip/hip_runtime.h>
#include <hip/hip_bf16.h>

typedef __attribute__((ext_vector_type(2))) float v2f;
typedef __attribute__((ext_vector_type(4))) float v4f;
typedef __attribute__((ext_vector_type(8))) float v8f;
typedef __attribute__((ext_vector_type(4))) unsigned int v4u;
typedef __attribute__((ext_vector_type(4))) int v4i;
typedef __attribute__((ext_vector_type(8))) int v8i;

#if defined(__gfx1250__) && __has_builtin(__builtin_amdgcn_tensor_load_to_lds)
#define USE_TDM 1
#else
#define USE_TDM 0
#endif

namespace {

constexpr int S  = 2048;
constexpr int B  = 4;
constexpr int D  = 4096;
constexpr int H  = 32;
constexpr int HD = 128;          // head dim
constexpr int M  = S * B;        // 8192 GEMM rows (s*B+b flattened)
constexpr int K  = D;            // 4096 reduction depth

constexpr int BM = 128;          // block tile M
constexpr int BN = 128;          // block tile N
constexpr int KC = 32;           // K slice staged in LDS
constexpr int LDA = KC + 4;      // 36-float LDS row: conflict-free frag reads, 16B-aligned rows

#if USE_TDM
// Issue one TDM 2D tile load: 128 rows x 32 f32 from a row-major [rows_total, rowlen]
// tensor into LDS at lds_off, padding each 32-DWORD row with 4 DWORDs so the LDS
// layout lands on the same conflict-free 36-float stride the compute loop expects.
// D# bitfields per CDNA5 ISA ch.8 (group0 128b, group1 256b; groups 2/3 zero: 2D).
// This toolchain exposes the 6-operand builtin: (g0 v4u, g1 v8i, v4i, v4i, v8i, cpol).
__device__ __forceinline__ void tdm_load_tile(unsigned lds_off, const float* gaddr,
                                              unsigned rows_total, unsigned rowlen)
{
    unsigned long long ga = (unsigned long long)(uintptr_t)gaddr;
    v4u g0;
    g0.x = 1u;                                   // count=1 (valid user descriptor)
    g0.y = lds_off;                              // lds_addr (bytes)
    g0.z = (unsigned)ga;                         // global_addr[31:0]
    g0.w = (unsigned)(ga >> 32) | (2u << 30);    // global_addr[56:32] | type=2 ("image")
    v8i g1;
    g1[0] = (int)((2u << 16)                     // data_size = 4 bytes
                | (1u << 20)                     // pad_enable
                | (4u << 22)                     // pad_interval: every 32 DWORDs
                | (3u << 25));                   // pad_amount: 4 DWORDs
    g1[1] = (int)((rowlen & 0xFFFFu) << 16);                                   // dim0 lo
    g1[2] = (int)(((rowlen >> 16) & 0xFFFFu) | ((rows_total & 0xFFFFu) << 16));// dim0 hi|dim1 lo
    g1[3] = (int)(((rows_total >> 16) & 0xFFFFu) | ((unsigned)KC << 16));      // dim1 hi|tile0=32
    g1[4] = (int)BM;                             // tile_dim1 = 128 rows, tile_dim2 = 0
    g1[5] = (int)rowlen;                         // tensor_dim0_stride[31:0] (elements)
    g1[6] = 0;                                   // stride0 hi | stride1 lo (2D: unused)
    g1[7] = 0;
    v4i z4 = {0, 0, 0, 0};
    v8i z8 = {0, 0, 0, 0, 0, 0, 0, 0};
    __builtin_amdgcn_tensor_load_to_lds(g0, g1, z4, z4, z8, 0);
}
#endif

// C[m,n] = sum_k X[m,k]*W[n,k] + bias[n]; optional fused RoPE epilogue (GEMM2).
// 8 waves: wm = wave&3 (32-row slice), wn = wave>>2 (64-col slice).
// Each wave: 2x4 grid of 16x16 f32 WMMA accumulators (wave tile 32x64).
// The 64-wide wave column span equals exactly one RoPE rotation group, so the
// rotate_half partner (channel c^32) is accumulator tile j^2, same lane/slot.
template <bool ROPE>
__global__ __launch_bounds__(256)
void gemm_wmma(const float* __restrict__ X, const float* __restrict__ W,
               const float* __restrict__ bias, float* __restrict__ out, int N,
               const int* __restrict__ pos_ids, const float* __restrict__ inv_freq)
{
    constexpr int BUF = (BM + BN) * LDA;         // floats per double-buffer slot
#if USE_TDM
    __shared__ float smem[2 * BUF];              // 72 KB: two [A|B] K-slices
#else
    __shared__ float smem[BUF];                  // 36 KB: single [A|B] K-slice
#endif

    const int tid   = threadIdx.x;
    const int lane  = tid & 31;
    const int wave  = tid >> 5;
    const int wm    = wave & 3;       // 0..3
    const int wn    = wave >> 2;      // 0..1
    const int l16   = lane & 15;
    const int lhalf = lane >> 4;      // 0|1  (k-pair select in 16x4 fragments)

    const int blockN = blockIdx.x * BN;
    const int blockM = blockIdx.y * BM;

    v8f acc[2][4];
#pragma unroll
    for (int i = 0; i < 2; ++i)
#pragma unroll
        for (int j = 0; j < 4; ++j) { v8f z = {}; acc[i][j] = z; }

    // WMMA fragment MACs over one staged K slice (Ab/Bb: 36-float row stride).
    auto compute_slice = [&](const float* __restrict__ Ab, const float* __restrict__ Bb) {
#pragma unroll
        for (int ks = 0; ks < KC; ks += 4) {
            const int kk = ks + 2 * lhalf;       // 16x4 A / 4x16 B fragment layout
            v2f af[2], bf[4];
#pragma unroll
            for (int i = 0; i < 2; ++i)
                af[i] = *(const v2f*)&Ab[(wm * 32 + i * 16 + l16) * LDA + kk];
#pragma unroll
            for (int j = 0; j < 4; ++j)
                bf[j] = *(const v2f*)&Bb[(wn * 64 + j * 16 + l16) * LDA + kk];
#pragma unroll
            for (int i = 0; i < 2; ++i)
#pragma unroll
                for (int j = 0; j < 4; ++j)
                    acc[i][j] = __builtin_amdgcn_wmma_f32_16x16x4_f32(
                        false, af[i], false, bf[j], (short)0, acc[i][j], false, false);
        }
    };

#if USE_TDM
    // --- TDM double-buffered pipeline: wave0 drives the DMA, all waves compute ---
    const unsigned ldsBase = (unsigned)(uintptr_t)(void*)smem;
    const float* gAbase = X + (size_t)blockM * K;
    const float* gBbase = W + (size_t)blockN * K;

    if (wave == 0) {                             // prime buffer 0 (2 TDM ops in flight)
        tdm_load_tile(ldsBase, gAbase, (unsigned)M, (unsigned)K);
        tdm_load_tile(ldsBase + (unsigned)(BM * LDA) * 4u, gBbase, (unsigned)N, (unsigned)K);
    }
    int p = 0;
    for (int k0 = 0; k0 < K; k0 += KC, p ^= 1) {
        if (wave == 0) {
            if (k0 + KC < K) {                   // async-fill the other buffer
                const unsigned off = ldsBase + (unsigned)((p ^ 1) * BUF) * 4u;
                tdm_load_tile(off, gAbase + (k0 + KC), (unsigned)M, (unsigned)K);
                tdm_load_tile(off + (unsigned)(BM * LDA) * 4u, gBbase + (k0 + KC),
                              (unsigned)N, (unsigned)K);
                __builtin_amdgcn_s_wait_tensorcnt(2);   // current slice retired
            } else {
                __builtin_amdgcn_s_wait_tensorcnt(0);
            }
        }
        __syncthreads();                         // slice p visible to all waves
        compute_slice(smem + p * BUF, smem + p * BUF + BM * LDA);
        __syncthreads();                         // done reading before TDM rewrites p
    }
#else
    // --- Fallback: manual global->LDS staging with register double-buffer ---
    const int ldRow = tid >> 1;                  // 0..127
    const int ldCol = (tid & 1) * 16;            // 0 | 16 (64B segment per thread)
    const float* gA = X + (size_t)(blockM + ldRow) * K + ldCol;
    const float* gB = W + (size_t)(blockN + ldRow) * K + ldCol;
    float* sA = &smem[ldRow * LDA + ldCol];
    float* sB = &smem[BM * LDA + ldRow * LDA + ldCol];

    v4f ra[4], rb[4];
#pragma unroll
    for (int u = 0; u < 4; ++u) {
        ra[u] = *(const v4f*)(gA + u * 4);
        rb[u] = *(const v4f*)(gB + u * 4);
    }
    for (int k0 = 0; k0 < K; k0 += KC) {
#pragma unroll
        for (int u = 0; u < 4; ++u) {
            *(v4f*)(sA + u * 4) = ra[u];
            *(v4f*)(sB + u * 4) = rb[u];
        }
        __syncthreads();
        if (k0 + 2 * KC < K) {
            __builtin_prefetch(gA + k0 + 2 * KC, 0, 3);
            __builtin_prefetch(gB + k0 + 2 * KC, 0, 3);
        }
        if (k0 + KC < K) {
#pragma unroll
            for (int u = 0; u < 4; ++u) {
                ra[u] = *(const v4f*)(gA + (k0 + KC) + u * 4);
                rb[u] = *(const v4f*)(gB + (k0 + KC) + u * 4);
            }
        }
        compute_slice(smem, smem + BM * LDA);
        __syncthreads();
    }
#endif

    // ---- Epilogue -----------------------------------------------------------
    const int colBase = blockN + wn * 64;        // 64-aligned wave column base
    float bj[4];
#pragma unroll
    for (int j = 0; j < 4; ++j) bj[j] = bias[colBase + j * 16 + l16];

    if (!ROPE) {
        // GEMM1: h = acc + bias, row-major [M,N]
#pragma unroll
        for (int i = 0; i < 2; ++i)
#pragma unroll
            for (int r = 0; r < 8; ++r) {
                const int mrow = blockM + wm * 32 + i * 16 + r + 8 * lhalf;
                float* orow = out + (size_t)mrow * N;
#pragma unroll
                for (int j = 0; j < 4; ++j)
                    orow[colBase + j * 16 + l16] = acc[i][j][r] + bj[j];
            }
    } else {
        // GEMM2: per-head channel layout [q(128)|k(128)|v(128)] repeating every 384.
        const int within = colBase % 384;        // one of {0,64,128,192,256,320}
        const int head   = colBase / 384;
        const bool isV   = within >= 256;
        const bool useP  = ((within >> 6) & 1) == 0;  // even 64-group -> primary stream
        int part, chan;
        if (within < 128)      { part = 0; chan = within;       }   // q
        else if (within < 256) { part = 1; chan = within - 128; }   // k
        else                   { part = 2; chan = within - 256; }   // v
        float* obase   = out + (size_t)part * M * D;
        const int ocol = head * HD + chan + l16; // + j*16 per tile

        const float f0 = inv_freq[l16];          // freq idx = (channel%64)%32
        const float f1 = inv_freq[16 + l16];

#pragma unroll
        for (int i = 0; i < 2; ++i)
#pragma unroll
            for (int r = 0; r < 8; ++r) {
                const int mrow = blockM + wm * 32 + i * 16 + r + 8 * lhalf;
                float s0 = 0.f, c0 = 1.f, s1 = 0.f, c1 = 1.f;
                if (!isV) {
                    const int srow = mrow >> 2;  // s  (B == 4)
                    const int brow = mrow & 3;   // b
                    const float p = (float)pos_ids[brow * 2 * S + (useP ? 0 : S) + srow];
                    __sincosf(p * f0, &s0, &c0);
                    __sincosf(p * f1, &s1, &c1);
                }
                float* orow = obase + (size_t)mrow * D;
#pragma unroll
                for (int j = 0; j < 4; ++j) {
                    const float v = acc[i][j][r] + bj[j];
                    float o;
                    if (isV) {
                        o = v;
                    } else {
                        const float vp = acc[i][j ^ 2][r] + bj[j ^ 2];   // channel c^32
                        const float cs = (j & 1) ? c1 : c0;
                        const float sn = (j & 1) ? s1 : s0;
                        o = (j < 2) ? (v * cs - vp * sn)                 // x1: -x2*sin
                                    : (v * cs + vp * sn);                // x2: +x1*sin
                    }
                    orow[ocol + j * 16] = o;
                }
            }
    }
}

} // namespace

extern "C" void kernel_launch(void* const* d_in, const int* in_sizes, int n_in,
                              void* d_out, int out_size, void* d_ws, size_t ws_size,
                              hipStream_t stream) {
    const float* xs      = (const float*)d_in[0];   // [S,B,D]
    const int*   pos_ids = (const int*)  d_in[1];   // [B,2,S]
    const float* W_aff   = (const float*)d_in[2];   // [D,D]
    const float* b_aff   = (const float*)d_in[3];   // [D]
    const float* W_qkv   = (const float*)d_in[4];   // [3D,D]
    const float* b_qkv   = (const float*)d_in[5];   // [3D]
    const float* inv_frq = (const float*)d_in[6];   // [32]

    float* h   = (float*)d_ws;                      // [8192,4096] fp32 = 128 MB scratch
    float* out = (float*)d_out;                     // q|k|v, each [8192,4096] fp32

    dim3 blk(256);
    dim3 g1(D / BN, M / BM);                        // 32 x 64
    gemm_wmma<false><<<g1, blk, 0, stream>>>(xs, W_aff, b_aff, h, D, nullptr, nullptr);
    dim3 g2(3 * D / BN, M / BM);                    // 96 x 64
    gemm_wmma<true><<<g2, blk, 0, stream>>>(h, W_qkv, b_qkv, out, 3 * D, pos_ids, inv_frq);
}